// PersistentTRMLP_1408749273827
// MI455X (gfx1250) — compile-verified
//
#include <hip/hip_runtime.h>
#include <math.h>

#define T_TOK 8192
#define HID   1024
#define INTER 4096
#define NEXP  8
#define BM    64      // tokens per workgroup tile (4 m-subtiles of 16)
#define BK1   32      // hidden (k1) chunk staged in LDS (double-buffered, async)
#define NCH1  (HID / BK1)
#define BK2   128     // inter (k2) chunk; 8 waves x 16 cols
#define ESPLIT 2      // experts split across gridDim.y (partial sums via atomics)
#define STR   162     // LDS pair-row stride in words (2*BM + 34 pad; 162 % 64 == 34)

typedef float v2f __attribute__((ext_vector_type(2)));
typedef float v8f __attribute__((ext_vector_type(8)));
typedef __attribute__((address_space(3))) float lds_f;

static __device__ __forceinline__ v8f wmma_f32(v2f a, v2f b, v8f c) {
  // D = A(16x4 f32) * B(4x16 f32) + C(16x16 f32)
  return __builtin_amdgcn_wmma_f32_16x16x4_f32(
      false, a, false, b, (short)0, c, false, false);
}

static __device__ __forceinline__ unsigned lds_off(const float* p) {
  return (unsigned)(unsigned long long)(lds_f*)p;  // group-segment byte offset
}

// async DMA: 8 bytes global -> LDS, tracked by ASYNCcnt (no VGPR staging)
static __device__ __forceinline__ void async_ld_b64(unsigned lds_byte_off,
                                                    const float* gptr) {
  asm volatile("global_load_async_to_lds_b64 %0, %1, off"
               :: "v"(lds_byte_off), "v"(gptr)
               : "memory");
}

static __device__ __forceinline__ void wait_async0() {
  asm volatile("s_wait_asynccnt 0x0" ::: "memory");
}

// ---------------------------------------------------------------------------
// Router: logits -> softmax -> top-2 -> renormalized combine weights [T, E],
// plus per-expert top-1 counts and prob sums for the aux loss.
// ---------------------------------------------------------------------------
__global__ __launch_bounds__(256) void moe_router_kernel(
    const float* __restrict__ x, const float* __restrict__ rw,
    float* __restrict__ combine, float* __restrict__ probsum,
    float* __restrict__ cnt) {
  const int t   = threadIdx.x;
  const int tok = blockIdx.x * 32 + (t >> 3);
  const int e   = t & 7;

  float acc = 0.f;
  const float* xr = x + (size_t)tok * HID;
  const float* wr = rw + (size_t)e * HID;
  for (int i = 0; i < HID; ++i) acc += xr[i] * wr[i];

  __shared__ float lg[32][9];
  lg[t >> 3][e] = acc;
  __syncthreads();

  if (e == 0) {
    const int row = t >> 3;
    float p[NEXP];
    float mx = lg[row][0];
    #pragma unroll
    for (int i = 1; i < NEXP; ++i) mx = fmaxf(mx, lg[row][i]);
    float s = 0.f;
    #pragma unroll
    for (int i = 0; i < NEXP; ++i) { p[i] = __expf(lg[row][i] - mx); s += p[i]; }
    const float inv = 1.f / s;
    #pragma unroll
    for (int i = 0; i < NEXP; ++i) p[i] *= inv;

    int i1 = 0;
    #pragma unroll
    for (int i = 1; i < NEXP; ++i) if (p[i] > p[i1]) i1 = i;
    int i2 = (i1 == 0) ? 1 : 0;
    #pragma unroll
    for (int i = 0; i < NEXP; ++i) if (i != i1 && p[i] > p[i2]) i2 = i;

    const float rs = 1.f / (p[i1] + p[i2]);
    float cw[NEXP];
    #pragma unroll
    for (int i = 0; i < NEXP; ++i) cw[i] = 0.f;
    cw[i1] = p[i1] * rs;
    cw[i2] = p[i2] * rs;
    #pragma unroll
    for (int i = 0; i < NEXP; ++i) combine[(size_t)tok * NEXP + i] = cw[i];

    atomicAdd(&cnt[i1], 1.0f);
    #pragma unroll
    for (int i = 0; i < NEXP; ++i) atomicAdd(&probsum[i], p[i]);
  }
}

__global__ void moe_zero_stats_kernel(float* __restrict__ stats) {
  if (threadIdx.x < 16) stats[threadIdx.x] = 0.f;
}

__global__ __launch_bounds__(256) void moe_zero_out_kernel(float* __restrict__ out) {
  const size_t i = ((size_t)blockIdx.x * 256 + threadIdx.x) * 4;
  *(float4*)(out + i) = make_float4(0.f, 0.f, 0.f, 0.f);
}

__global__ void moe_aux_kernel(const float* __restrict__ probsum,
                               const float* __restrict__ cnt,
                               float* __restrict__ aux) {
  if (threadIdx.x == 0) {
    float a = 0.f;
    #pragma unroll
    for (int i = 0; i < NEXP; ++i)
      a += (cnt[i] / (float)T_TOK) * (probsum[i] / (float)T_TOK);
    *aux = a * (float)NEXP;
  }
}

// ---------------------------------------------------------------------------
// Fused expert MLP. Per 64-token tile, loop 4 experts (grid.y splits experts),
// keep the silu'd h chunk (combine weight folded in) in LDS, accumulate the
// 64x1024 output tile in registers, atomically add partials at the end.
// x chunks are staged memory->LDS with GLOBAL_LOAD_ASYNC_TO_LDS_B64 into a
// double buffer so staging overlaps the WMMA stream; pair-major layouts give
// single-b64 WMMA A fragments.
// ---------------------------------------------------------------------------
__global__ __launch_bounds__(256) void moe_mlp_kernel(
    const float* __restrict__ x, const float* __restrict__ w1,
    const float* __restrict__ w2, const float* __restrict__ combine,
    float* __restrict__ out) {
  __shared__ float xAb[2][(BK1 / 2) * STR];  // x chunks, pair-major [k1/2][m][2]
  __shared__ float hbA[(BK2 / 2) * STR];     // silu(h)*comb chunk, pair-major
  __shared__ float comb[BM];

  const int tok0 = blockIdx.x * BM;
  const int lane = threadIdx.x & 31;
  const int wid  = threadIdx.x >> 5;         // 8 waves
  const int half = lane >> 4;                // selects K0/K1 vs K2/K3 fragments
  const int lm   = lane & 15;

  // per-thread staging coordinates: 1024 pairs per chunk, 4 per thread
  const int skp[1] = {0};
  (void)skp;

  v8f c[4][8];                               // 4 m-subtiles x 8 n-subtiles
  #pragma unroll
  for (int mt = 0; mt < 4; ++mt)
    #pragma unroll
    for (int nt = 0; nt < 8; ++nt) c[mt][nt] = (v8f)0.f;

  const unsigned xbase0 = lds_off(&xAb[0][0]);
  const unsigned xbase1 = lds_off(&xAb[1][0]);

  const int e0 = blockIdx.y * (NEXP / ESPLIT);
  for (int e = e0; e < e0 + NEXP / ESPLIT; ++e) {
    if (threadIdx.x < BM)
      comb[threadIdx.x] = combine[(size_t)(tok0 + threadIdx.x) * NEXP + e];
    __syncthreads();

    const float* W1 = w1 + (size_t)e * HID * INTER;
    const float* W2 = w2 + (size_t)e * INTER * HID;

    for (int k2 = 0; k2 < INTER; k2 += BK2) {
      // ---- GEMM1: h(wave's 16 cols) = x_tile @ W1[:, k2+wid*16 .. +16) ----
      v8f h[4];
      #pragma unroll
      for (int mt = 0; mt < 4; ++mt) h[mt] = (v8f)0.f;

      // prologue: async-stage chunk 0 into buffer 0
      #pragma unroll
      for (int r = 0; r < 4; ++r) {
        const int q  = threadIdx.x + 256 * r;
        const int kp = q & 15;
        const int m  = q >> 4;
        async_ld_b64(xbase0 + (unsigned)((kp * STR + 2 * m) * 4),
                     x + (size_t)(tok0 + m) * HID + 2 * kp);
      }

      #pragma unroll 2
      for (int ic = 0; ic < NCH1; ++ic) {
        wait_async0();          // own async writes done
        __syncthreads();        // everyone's async writes done
        if (ic + 1 < NCH1) {
          const unsigned nb = ((ic + 1) & 1) ? xbase1 : xbase0;
          const int nk1 = (ic + 1) * BK1;
          #pragma unroll
          for (int r = 0; r < 4; ++r) {
            const int q  = threadIdx.x + 256 * r;
            const int kp = q & 15;
            const int m  = q >> 4;
            async_ld_b64(nb + (unsigned)((kp * STR + 2 * m) * 4),
                         x + (size_t)(tok0 + m) * HID + nk1 + 2 * kp);
          }
        }

        const float* xbuf = xAb[ic & 1];
        const int k1 = ic * BK1;
        const int colW1 = k2 + wid * 16 + lm;
        #pragma unroll
        for (int kk = 0; kk < BK1; kk += 4) {
          const float* ap = xbuf + ((kk >> 1) + half) * STR + 2 * lm;
          const v2f a0 = *(const v2f*)(ap);
          const v2f a1 = *(const v2f*)(ap + 32);
          const v2f a2 = *(const v2f*)(ap + 64);
          const v2f a3 = *(const v2f*)(ap + 96);
          const float* bp = W1 + (size_t)(k1 + kk + 2 * half) * INTER + colW1;
          v2f b;
          b.x = bp[0];
          b.y = bp[INTER];
          h[0] = wmma_f32(a0, b, h[0]);
          h[1] = wmma_f32(a1, b, h[1]);
          h[2] = wmma_f32(a2, b, h[2]);
          h[3] = wmma_f32(a3, b, h[3]);
        }
      }

      // ---- silu (fast rcp) + fold combine weight, stash h chunk in LDS ----
      {
        const int hcol = wid * 16 + lm;
        float* hw = hbA + (hcol >> 1) * STR + (hcol & 1);
        #pragma unroll
        for (int mt = 0; mt < 4; ++mt)
          #pragma unroll
          for (int j = 0; j < 8; ++j) {
            const int m = mt * 16 + j + 8 * half;
            const float v = h[mt][j];
            const float s = v * __builtin_amdgcn_rcpf(1.f + __expf(-v));
            hw[2 * m] = s * comb[m];
          }
      }
      __syncthreads();

      // ---- GEMM2: c += h_chunk(64x128) @ W2[k2 chunk, wave's 128 cols] ----
      const int colW2 = wid * 128 + lm;
      #pragma unroll 2
      for (int kk = 0; kk < BK2; kk += 4) {
        const float* ap = hbA + ((kk >> 1) + half) * STR + 2 * lm;
        const v2f a0 = *(const v2f*)(ap);
        const v2f a1 = *(const v2f*)(ap + 32);
        const v2f a2 = *(const v2f*)(ap + 64);
        const v2f a3 = *(const v2f*)(ap + 96);
        const float* bp = W2 + (size_t)(k2 + kk + 2 * half) * HID + colW2;
        #pragma unroll
        for (int nt = 0; nt < 8; ++nt) {
          v2f b;
          b.x = bp[nt * 16];
          b.y = bp[nt * 16 + HID];
          c[0][nt] = wmma_f32(a0, b, c[0][nt]);
          c[1][nt] = wmma_f32(a1, b, c[1][nt]);
          c[2][nt] = wmma_f32(a2, b, c[2][nt]);
          c[3][nt] = wmma_f32(a3, b, c[3][nt]);
        }
      }
      __syncthreads();
    }
  }

  // ---- accumulate partial output tile (summed over this WG's experts) ----
  #pragma unroll
  for (int mt = 0; mt < 4; ++mt)
    #pragma unroll
    for (int nt = 0; nt < 8; ++nt)
      #pragma unroll
      for (int j = 0; j < 8; ++j) {
        const int m = mt * 16 + j + 8 * half;
        atomicAdd(&out[(size_t)(tok0 + m) * HID + wid * 128 + nt * 16 + lm],
                  c[mt][nt][j]);
      }
}

// ---------------------------------------------------------------------------
extern "C" void kernel_launch(void* const* d_in, const int* in_sizes, int n_in,
                              void* d_out, int out_size, void* d_ws, size_t ws_size,
                              hipStream_t stream) {
  (void)in_sizes; (void)n_in; (void)out_size; (void)ws_size;
  const float* x  = (const float*)d_in[0];
  const float* rw = (const float*)d_in[1];
  const float* w1 = (const float*)d_in[2];
  const float* w2 = (const float*)d_in[3];
  float* out = (float*)d_out;
  float* ws  = (float*)d_ws;

  float* combine = ws;                        // T*E floats (256 KB)
  float* stats   = ws + (size_t)T_TOK * NEXP; // probsum[8] + cnt[8]
  float* probsum = stats;
  float* cnt     = stats + NEXP;
  float* aux_out = out + (size_t)T_TOK * HID;

  moe_zero_stats_kernel<<<1, 32, 0, stream>>>(stats);
  moe_zero_out_kernel<<<(T_TOK * HID) / (256 * 4), 256, 0, stream>>>(out);
  moe_router_kernel<<<T_TOK / 32, 256, 0, stream>>>(x, rw, combine, probsum, cnt);
  moe_aux_kernel<<<1, 32, 0, stream>>>(probsum, cnt, aux_out);

  dim3 grid(T_TOK / BM, ESPLIT, 1);
  moe_mlp_kernel<<<grid, 256, 0, stream>>>(x, w1, w2, combine, out);
}